// GraphConvolution_38242388803691
// MI455X (gfx1250) — compile-verified
//
#include <hip/hip_runtime.h>

// ---------------------------------------------------------------------------
// GCN layer for MI455X (gfx1250, wave32):
//   pre  = x @ W                       (fp32 WMMA 16x16x4, full precision)
//   out  = A_sparse @ pre              (sorted-COO run-length SpMM, L2-resident gather)
//   out  = relu((out - mean) * rsqrt(var + eps))   (column-wise moments)
// ---------------------------------------------------------------------------

typedef float v2f __attribute__((ext_vector_type(2)));
typedef float v8f __attribute__((ext_vector_type(8)));

#define F_IN  256
#define F_OUT 128
#define EPS_BN 1e-3f
#define EPB 256   // edges per block in SpMM

// ---------------------------------------------------------------------------
// Kernel 1: pre[M,128] = x[M,256] @ W[256,128] using V_WMMA_F32_16X16X4_F32.
// One wave per 16-row M-tile; each wave computes all 8 column tiles (8 x v8f
// accumulators) so the A fragment is reused 8x per K-step. 256 threads = 8
// waves = 8 M-tiles per block.
//
// fp32 WMMA fragment layout (wave32):
//   A (16x4): lane<16 -> row M=lane,  VGPR0/1 = K{0,1}; lane>=16 -> K{2,3}
//   B (4x16): lane<16 -> col N=lane,  VGPR0/1 = K{0,1}; lane>=16 -> K{2,3}
//   C/D     : VGPR r -> row r + 8*(lane>=16), col = lane&15
// ---------------------------------------------------------------------------
__global__ __launch_bounds__(256) void gemm_wmma_f32(const float* __restrict__ x,
                                                     const float* __restrict__ W,
                                                     float* __restrict__ pre,
                                                     int M) {
    const int lane = threadIdx.x & 31;
    const int wave = threadIdx.x >> 5;
    const int numTiles = (M + 15) >> 4;
    const int tile = blockIdx.x * 8 + wave;
    if (tile >= numTiles) return;          // uniform per wave: EXEC stays all-1s

    const int m0   = tile << 4;
    const int half = lane >> 4;            // 0 -> K pair {0,1}, 1 -> K pair {2,3}
    const int l15  = lane & 15;

    int arow = m0 + l15;
    if (arow >= M) arow = M - 1;           // clamp loads; stores are guarded below
    const float* aptr = x + (size_t)arow * F_IN + 2 * half;

    v8f acc[8] = {};                       // 8 N-tiles x (16x16 f32 / 32 lanes)

    for (int k0 = 0; k0 < F_IN; k0 += 4) {
        const v2f a = *(const v2f*)(aptr + k0);   // global_load_b64
        const int kb = k0 + 2 * half;
        #pragma unroll
        for (int t = 0; t < 8; ++t) {
            const int col = (t << 4) + l15;
            v2f b;
            b.x = W[(size_t)kb * F_OUT + col];
            b.y = W[(size_t)(kb + 1) * F_OUT + col];
            acc[t] = __builtin_amdgcn_wmma_f32_16x16x4_f32(
                false, a, false, b, (short)0, acc[t], false, false);
        }
    }

    #pragma unroll
    for (int t = 0; t < 8; ++t) {
        const int col = (t << 4) + l15;
        #pragma unroll
        for (int r = 0; r < 8; ++r) {
            const int row = m0 + r + 8 * half;
            if (row < M) pre[(size_t)row * F_OUT + col] = acc[t][r];
        }
    }
}

// ---------------------------------------------------------------------------
// Kernel 2: SpMM with sorted-row run-length accumulation.
// Block = 128 threads, one feature column each. Edge metadata is staged in
// LDS once per block (scalar-uniform across the block), then each thread
// accumulates vals[e]*pre[cols[e]][f] in a register while the destination
// row stays constant, flushing one atomicAdd per (row-run, feature).
// `pre` (51 MB) is L2-resident -> the gather runs at L2 bandwidth.
// ---------------------------------------------------------------------------
__global__ __launch_bounds__(128) void spmm_sorted(const float* __restrict__ pre,
                                                   const float* __restrict__ vals,
                                                   const int* __restrict__ rows,
                                                   const int* __restrict__ cols,
                                                   float* __restrict__ out,
                                                   int E) {
    __shared__ int   s_row[EPB];
    __shared__ int   s_col[EPB];
    __shared__ float s_val[EPB];

    const int f  = threadIdx.x;            // feature 0..127
    const int e0 = blockIdx.x * EPB;
    const int n  = (E - e0 < EPB) ? (E - e0) : EPB;
    if (n <= 0) return;

    for (int i = threadIdx.x; i < n; i += 128) {
        s_row[i] = rows[e0 + i];
        s_col[i] = cols[e0 + i];
        s_val[i] = vals[e0 + i];
    }
    __syncthreads();

    int   cur = s_row[0];
    float acc = 0.f;
    for (int i = 0; i < n; ++i) {
        const int r = s_row[i];            // uniform branch: scalar control flow
        if (r != cur) {
            atomicAdd(out + (size_t)cur * F_OUT + f, acc);
            acc = 0.f;
            cur = r;
        }
        acc = fmaf(s_val[i], pre[(size_t)s_col[i] * F_OUT + f], acc);
    }
    atomicAdd(out + (size_t)cur * F_OUT + f, acc);
}

// ---------------------------------------------------------------------------
// Kernel 3: per-feature sum / sum-of-squares partial reduction.
// Block = 128 threads (one feature each) over a contiguous row range;
// one atomicAdd per block per stat.
// ---------------------------------------------------------------------------
__global__ __launch_bounds__(128) void col_stats(const float* __restrict__ out,
                                                 float* __restrict__ st,
                                                 int Nrows, int rowsPerBlock) {
    const int f  = threadIdx.x;
    const int r0 = blockIdx.x * rowsPerBlock;
    int r1 = r0 + rowsPerBlock;
    if (r1 > Nrows) r1 = Nrows;

    float s = 0.f, s2 = 0.f;
    for (int r = r0; r < r1; ++r) {
        const float v = out[(size_t)r * F_OUT + f];
        s  += v;
        s2  = fmaf(v, v, s2);
    }
    atomicAdd(st + f, s);
    atomicAdd(st + F_OUT + f, s2);
}

// ---------------------------------------------------------------------------
// Kernel 4: batch-norm (no scale/offset) + ReLU, elementwise in-place.
// ---------------------------------------------------------------------------
__global__ __launch_bounds__(256) void bn_relu(float* __restrict__ out,
                                               const float* __restrict__ st,
                                               long total, float invN) {
    long i = (long)blockIdx.x * blockDim.x + threadIdx.x;
    const long stride = (long)gridDim.x * blockDim.x;
    for (; i < total; i += stride) {
        const int   f    = (int)(i & (F_OUT - 1));
        const float mean = st[f] * invN;
        const float var  = fmaf(-mean, mean, st[F_OUT + f] * invN);
        const float v    = (out[i] - mean) * rsqrtf(var + EPS_BN);
        out[i] = fmaxf(v, 0.f);
    }
}

// ---------------------------------------------------------------------------
// Launcher. Workspace layout: [0,1KB) = 256 stat floats, [1KB, ...) = pre.
// d_out is zeroed every call (it is both the SpMM accumulator and the final
// result buffer); memsets are async on `stream` so graph capture is legal.
// ---------------------------------------------------------------------------
extern "C" void kernel_launch(void* const* d_in, const int* in_sizes, int n_in,
                              void* d_out, int out_size, void* d_ws, size_t ws_size,
                              hipStream_t stream) {
    const float* x    = (const float*)d_in[0];
    const float* W    = (const float*)d_in[1];
    const float* vals = (const float*)d_in[2];
    const int*   rows = (const int*)d_in[3];
    const int*   cols = (const int*)d_in[4];

    const int M = in_sizes[0] / F_IN;   // 100000
    const int E = in_sizes[2];          // 3200000

    float* st  = (float*)d_ws;                       // 2*128 floats
    float* pre = (float*)((char*)d_ws + 1024);       // M*128 floats

    hipMemsetAsync(d_out, 0, (size_t)out_size * sizeof(float), stream);
    hipMemsetAsync(st, 0, 2 * F_OUT * sizeof(float), stream);

    {   // GEMM: 8 M-tiles (waves) per 256-thread block
        const int numTiles = (M + 15) / 16;
        const int blocks   = (numTiles + 7) / 8;
        gemm_wmma_f32<<<blocks, 256, 0, stream>>>(x, W, pre, M);
    }
    {   // SpMM over sorted COO
        const int blocks = (E + EPB - 1) / EPB;
        spmm_sorted<<<blocks, 128, 0, stream>>>(pre, vals, rows, cols, (float*)d_out, E);
    }
    {   // column moments
        const int blocks = 256;
        const int rpb    = (M + blocks - 1) / blocks;
        col_stats<<<blocks, 128, 0, stream>>>((const float*)d_out, st, M, rpb);
    }
    {   // normalize + relu
        const long total = (long)M * F_OUT;
        bn_relu<<<2048, 256, 0, stream>>>((float*)d_out, st, total, 1.0f / (float)M);
    }
}